// DeePC_70909910057710
// MI455X (gfx1250) — compile-verified
//
#include <hip/hip_runtime.h>

// ---- problem dimensions (T=3000, M=8, P=12, TINI=20, N=40, NB=128) ----
#define G      2941   // g_dim (hankel columns)
#define GP     2944   // G padded to 16
#define NK     3341   // KKT dim (G + 400)
#define NKP    3344   // padded KKT dim
#define NTK    209    // NKP/16 row tiles
#define NBATCH 128    // RHS columns
#define WFULL  3472   // NKP + NBATCH (augmented width)
#define WT     217    // WFULL/16
#define RHS0   3344   // first rhs column in Kaug
#define KS     800    // rows of S = [Yf(480); Uf(320)]
#define PW     3072   // GP + NBATCH (product width of the H/rhs GEMM)

typedef __attribute__((ext_vector_type(2))) float v2f;
typedef __attribute__((ext_vector_type(8))) float v8f;

__device__ __forceinline__ v8f wmma4(v2f a, v2f b, v8f c) {
  // D = A(16x4) * B(4x16) + C(16x16), fp32 matrix path on gfx1250
  return __builtin_amdgcn_wmma_f32_16x16x4_f32(false, a, false, b, (short)0, c,
                                               false, false);
}

// ---------------------------------------------------------------------------
// K0: build S (800 x GP, zero-padded) and SWe = [diag(w)*S | Z] (800 x PW)
//     Z[k][b] = w[k] * (k<480 ? yref[b][k] : uref[b][k-480])
// ---------------------------------------------------------------------------
__global__ void build_S(const float* __restrict__ Yf, const float* __restrict__ Uf,
                        const float* __restrict__ q, const float* __restrict__ r,
                        const float* __restrict__ yref, const float* __restrict__ uref,
                        float* __restrict__ Sb, float* __restrict__ SWe) {
  int idx = blockIdx.x * blockDim.x + threadIdx.x;
  if (idx >= KS * PW) return;
  int k = idx / PW, j = idx % PW;
  float w = (k < 480) ? q[k % 12] : r[(k - 480) % 8];
  if (j < GP) {
    float s = 0.0f;
    if (j < G) s = (k < 480) ? Yf[k * G + j] : Uf[(k - 480) * G + j];
    Sb[k * GP + j]  = s;
    SWe[k * PW + j] = w * s;
  } else {
    int b = j - GP;
    float z = (k < 480) ? yref[b * 480 + k] : uref[b * 320 + (k - 480)];
    SWe[k * PW + j] = w * z;
  }
}

// ---------------------------------------------------------------------------
// K1: P = S^T * SWe via v_wmma_f32_16x16x4_f32; fused epilogue writes
//     Kaug[i][j] = 2*(P + lam*IPI)   (i<G, j<G)
//     Kaug[i][RHS0+b] = 2*P          (i<G, rhs cols)
// Each wave computes a 16x64 strip (A operand reused across 4 N-tiles).
// ---------------------------------------------------------------------------
__global__ void gemm_H_rhs(const float* __restrict__ Sb, const float* __restrict__ SWe,
                           const float* __restrict__ IPI, const float* __restrict__ lam_p,
                           float* __restrict__ Kaug) {
  int wave = (blockIdx.x * blockDim.x + threadIdx.x) >> 5;   // wave-uniform
  int lane = threadIdx.x & 31;
  const int NSTRIP = PW / 64;                                 // 48
  const int MT = GP / 16;                                     // 184
  if (wave >= MT * NSTRIP) return;
  int ti = wave / NSTRIP, str = wave % NSTRIP;
  int half = lane >> 4, lm = lane & 15;
  int i0 = ti * 16, j0 = str * 64;
  v8f acc0 = {0,0,0,0,0,0,0,0}, acc1 = acc0, acc2 = acc0, acc3 = acc0;
  for (int k = 0; k < KS; k += 4) {
    int ka = k + 2 * half;
    v2f a;
    a[0] = Sb[(ka + 0) * GP + i0 + lm];   // A = S^T : A[m][kk] = S[kk][i0+m]
    a[1] = Sb[(ka + 1) * GP + i0 + lm];
    v2f b;
    b[0] = SWe[(ka + 0) * PW + j0 +  0 + lm];
    b[1] = SWe[(ka + 1) * PW + j0 +  0 + lm];
    acc0 = wmma4(a, b, acc0);
    b[0] = SWe[(ka + 0) * PW + j0 + 16 + lm];
    b[1] = SWe[(ka + 1) * PW + j0 + 16 + lm];
    acc1 = wmma4(a, b, acc1);
    b[0] = SWe[(ka + 0) * PW + j0 + 32 + lm];
    b[1] = SWe[(ka + 1) * PW + j0 + 32 + lm];
    acc2 = wmma4(a, b, acc2);
    b[0] = SWe[(ka + 0) * PW + j0 + 48 + lm];
    b[1] = SWe[(ka + 1) * PW + j0 + 48 + lm];
    acc3 = wmma4(a, b, acc3);
  }
  float lam = lam_p[0];
  v8f accs[4] = {acc0, acc1, acc2, acc3};
  for (int t = 0; t < 4; ++t) {
    int jb = j0 + t * 16 + lm;
#pragma unroll
    for (int rr = 0; rr < 8; ++rr) {
      int i = i0 + rr + 8 * half;
      if (i >= G) continue;
      float v = accs[t][rr];
      if (jb < G) {
        Kaug[(size_t)i * WFULL + jb] = 2.0f * (v + lam * IPI[(size_t)i * G + jb]);
      } else if (jb >= GP) {
        Kaug[(size_t)i * WFULL + RHS0 + (jb - GP)] = 2.0f * v;
      }
    }
  }
}

// ---------------------------------------------------------------------------
// K2: fill the rest of the augmented matrix [K | rhs] (regions K1 skips)
// ---------------------------------------------------------------------------
__global__ void assemble_K(const float* __restrict__ Up, const float* __restrict__ Yp,
                           const float* __restrict__ u_ini, const float* __restrict__ y_ini,
                           float* __restrict__ Kaug) {
  int idx = blockIdx.x * blockDim.x + threadIdx.x;
  if (idx >= NKP * WFULL) return;
  int i = idx / WFULL, j = idx % WFULL;
  if (i < G) {
    if (j < G || j >= RHS0) return;            // written by gemm_H_rhs
    float v = 0.0f;
    if (j < NK) {                               // A^T block: K[i][G+a] = A[a][i]
      int a = j - G;
      v = (a < 160) ? Up[(size_t)a * G + i] : Yp[(size_t)(a - 160) * G + i];
    }
    Kaug[(size_t)i * WFULL + j] = v;
  } else if (i < NK) {
    int a = i - G;
    float v = 0.0f;
    if (j < G) {
      v = (a < 160) ? Up[(size_t)a * G + j] : Yp[(size_t)(a - 160) * G + j];
    } else if (j >= RHS0) {
      int b = j - RHS0;
      v = (a < 160) ? u_ini[b * 160 + a] : y_ini[b * 240 + (a - 160)];
    }
    Kaug[(size_t)i * WFULL + j] = v;
  } else {                                      // padding rows 3341..3343: identity
    Kaug[(size_t)i * WFULL + j] = (i == j) ? 1.0f : 0.0f;
  }
}

// ---------------------------------------------------------------------------
// LU step k: diag 16x16 LU (no pivot; KKT is quasi-definite)
// ---------------------------------------------------------------------------
__global__ void lu_diag(float* __restrict__ Kaug, int k) {
  __shared__ float A[16][17];
  __shared__ float lcol[16];
  int rr = threadIdx.x >> 4, c = threadIdx.x & 15;
  A[rr][c] = Kaug[(size_t)(k * 16 + rr) * WFULL + k * 16 + c];
  __syncthreads();
  for (int s = 0; s < 15; ++s) {
    if (c == s && rr > s) lcol[rr] = A[rr][s] / A[s][s];
    __syncthreads();
    if (rr > s) {
      if (c > s)       A[rr][c] -= lcol[rr] * A[s][c];
      else if (c == s) A[rr][c]  = lcol[rr];
    }
    __syncthreads();
  }
  Kaug[(size_t)(k * 16 + rr) * WFULL + k * 16 + c] = A[rr][c];
}

// LU step k: L21 = A21 * U11^{-1}, one thread per row below the diagonal block
__global__ void lu_l21(float* __restrict__ Kaug, int k) {
  __shared__ float U[16][17];
  int tid = threadIdx.x;
  { int rr = tid >> 4, c = tid & 15;
    U[rr][c] = Kaug[(size_t)(k * 16 + rr) * WFULL + k * 16 + c]; }
  __syncthreads();
  int row = (k + 1) * 16 + blockIdx.x * blockDim.x + tid;
  if (row >= NKP) return;
  float* p = Kaug + (size_t)row * WFULL + k * 16;
  float a[16];
#pragma unroll
  for (int j = 0; j < 16; ++j) a[j] = p[j];
#pragma unroll
  for (int j = 0; j < 16; ++j) {
    float v = a[j];
    for (int t = 0; t < j; ++t) v -= a[t] * U[t][j];
    a[j] = v / U[j][j];
  }
#pragma unroll
  for (int j = 0; j < 16; ++j) p[j] = a[j];
}

// LU step k: U12 = L11^{-1} * A12 (includes rhs columns => forward subst. free)
__global__ void lu_u12(float* __restrict__ Kaug, int k) {
  __shared__ float Lt[16][17];
  int tid = threadIdx.x;
  { int rr = tid >> 4, c = tid & 15;
    Lt[rr][c] = Kaug[(size_t)(k * 16 + rr) * WFULL + k * 16 + c]; }
  __syncthreads();
  int col = (k + 1) * 16 + blockIdx.x * blockDim.x + tid;
  if (col >= WFULL) return;
  float a[16];
#pragma unroll
  for (int i = 0; i < 16; ++i) a[i] = Kaug[(size_t)(k * 16 + i) * WFULL + col];
#pragma unroll
  for (int i = 1; i < 16; ++i) {
    float v = a[i];
    for (int t = 0; t < i; ++t) v -= Lt[i][t] * a[t];
    a[i] = v;
  }
#pragma unroll
  for (int i = 0; i < 16; ++i) Kaug[(size_t)(k * 16 + i) * WFULL + col] = a[i];
}

// LU step k: trailing update A22 -= L21*U12, one 16x16 tile per wave via WMMA
__global__ void lu_trailing(float* __restrict__ Kaug, int k) {
  int wave = (blockIdx.x * blockDim.x + threadIdx.x) >> 5;   // wave-uniform
  int lane = threadIdx.x & 31;
  int RT = (NTK - 1) - k;      // row tiles k+1..208
  int CT = (WT - 1) - k;       // col tiles k+1..216 (incl. rhs)
  if (wave >= RT * CT) return;
  int ti = k + 1 + wave / CT;
  int tj = k + 1 + wave % CT;
  int half = lane >> 4, lm = lane & 15;
  int r0 = ti * 16, c0 = tj * 16, kc = k * 16;
  v8f acc;
#pragma unroll
  for (int rr = 0; rr < 8; ++rr)
    acc[rr] = Kaug[(size_t)(r0 + rr + 8 * half) * WFULL + c0 + lm];
#pragma unroll
  for (int ks = 0; ks < 16; ks += 4) {
    int ka = kc + ks + 2 * half;
    v2f a, b;
    a[0] = -Kaug[(size_t)(r0 + lm) * WFULL + ka + 0];   // -L21 (fold negation)
    a[1] = -Kaug[(size_t)(r0 + lm) * WFULL + ka + 1];
    b[0] =  Kaug[(size_t)(ka + 0) * WFULL + c0 + lm];   // U12
    b[1] =  Kaug[(size_t)(ka + 1) * WFULL + c0 + lm];
    acc = wmma4(a, b, acc);
  }
#pragma unroll
  for (int rr = 0; rr < 8; ++rr)
    Kaug[(size_t)(r0 + rr + 8 * half) * WFULL + c0 + lm] = acc[rr];
}

// ---------------------------------------------------------------------------
// Back substitution: diag solve x = U11^{-1} y (128 rhs), in place in rhs cols
// ---------------------------------------------------------------------------
__global__ void bs_diag(float* __restrict__ Kaug, int kb) {
  __shared__ float U[16][17];
  int tid = threadIdx.x;  // 128 threads
  for (int t = tid; t < 256; t += 128)
    U[t >> 4][t & 15] = Kaug[(size_t)(kb * 16 + (t >> 4)) * WFULL + kb * 16 + (t & 15)];
  __syncthreads();
  if (tid < NBATCH) {
    float x[16];
#pragma unroll
    for (int i = 0; i < 16; ++i)
      x[i] = Kaug[(size_t)(kb * 16 + i) * WFULL + RHS0 + tid];
#pragma unroll
    for (int i = 15; i >= 0; --i) {
      float v = x[i];
      for (int j = i + 1; j < 16; ++j) v -= U[i][j] * x[j];
      x[i] = v / U[i][i];
    }
#pragma unroll
    for (int i = 0; i < 16; ++i)
      Kaug[(size_t)(kb * 16 + i) * WFULL + RHS0 + tid] = x[i];
  }
}

// Back substitution: y[above] -= U(:,kb)*x(kb), one 16x16 tile per wave (WMMA)
__global__ void bs_update(float* __restrict__ Kaug, int kb) {
  int wave = (blockIdx.x * blockDim.x + threadIdx.x) >> 5;
  int lane = threadIdx.x & 31;
  const int CT = NBATCH / 16;  // 8
  if (wave >= kb * CT) return;
  int ti = wave / CT, tc = wave % CT;
  int half = lane >> 4, lm = lane & 15;
  int r0 = ti * 16, kc = kb * 16, c0 = RHS0 + tc * 16;
  v8f acc;
#pragma unroll
  for (int rr = 0; rr < 8; ++rr)
    acc[rr] = Kaug[(size_t)(r0 + rr + 8 * half) * WFULL + c0 + lm];
#pragma unroll
  for (int ks = 0; ks < 16; ks += 4) {
    int ka = kc + ks + 2 * half;
    v2f a, b;
    a[0] = -Kaug[(size_t)(r0 + lm) * WFULL + ka + 0];   // -U(ti,kb)
    a[1] = -Kaug[(size_t)(r0 + lm) * WFULL + ka + 1];
    b[0] =  Kaug[(size_t)(ka + 0) * WFULL + c0 + lm];   // x(kb)
    b[1] =  Kaug[(size_t)(ka + 1) * WFULL + c0 + lm];
    acc = wmma4(a, b, acc);
  }
#pragma unroll
  for (int rr = 0; rr < 8; ++rr)
    Kaug[(size_t)(r0 + rr + 8 * half) * WFULL + c0 + lm] = acc[rr];
}

// Extract g (rows 0..G-1 of the solution) into zero-padded Xg (GP x 128)
__global__ void extract_g(const float* __restrict__ Kaug, float* __restrict__ Xg) {
  int idx = blockIdx.x * blockDim.x + threadIdx.x;
  if (idx >= GP * NBATCH) return;
  int i = idx / NBATCH, b = idx % NBATCH;
  Xg[idx] = (i < G) ? Kaug[(size_t)i * WFULL + RHS0 + b] : 0.0f;
}

// out[b][m] = sum_k S[srow0+m][k] * Xg[k][b]  (writes transposed result)
__global__ void out_gemm(const float* __restrict__ Sb, const float* __restrict__ Xg,
                         float* __restrict__ outp, int srow0, int MT, int outLD) {
  int wave = (blockIdx.x * blockDim.x + threadIdx.x) >> 5;
  int lane = threadIdx.x & 31;
  const int NT = NBATCH / 16;  // 8
  if (wave >= MT * NT) return;
  int ti = wave / NT, tn = wave % NT;
  int half = lane >> 4, lm = lane & 15;
  v8f acc = {0,0,0,0,0,0,0,0};
  for (int k = 0; k < GP; k += 4) {
    int ka = k + 2 * half;
    v2f a, b;
    a[0] = Sb[(size_t)(srow0 + ti * 16 + lm) * GP + ka + 0];
    a[1] = Sb[(size_t)(srow0 + ti * 16 + lm) * GP + ka + 1];
    b[0] = Xg[(ka + 0) * NBATCH + tn * 16 + lm];
    b[1] = Xg[(ka + 1) * NBATCH + tn * 16 + lm];
    acc = wmma4(a, b, acc);
  }
#pragma unroll
  for (int rr = 0; rr < 8; ++rr) {
    int m = ti * 16 + rr + 8 * half;
    int n = tn * 16 + lm;
    outp[(size_t)n * outLD + m] = acc[rr];
  }
}

// ---------------------------------------------------------------------------
extern "C" void kernel_launch(void* const* d_in, const int* in_sizes, int n_in,
                              void* d_out, int out_size, void* d_ws, size_t ws_size,
                              hipStream_t stream) {
  (void)in_sizes; (void)n_in; (void)out_size; (void)ws_size;
  const float* Up    = (const float*)d_in[0];
  const float* Yp    = (const float*)d_in[1];
  const float* Uf    = (const float*)d_in[2];
  const float* Yf    = (const float*)d_in[3];
  const float* IPI   = (const float*)d_in[4];
  const float* q     = (const float*)d_in[5];
  const float* r     = (const float*)d_in[6];
  const float* lam   = (const float*)d_in[7];
  const float* yref  = (const float*)d_in[8];
  const float* uref  = (const float*)d_in[9];
  const float* u_ini = (const float*)d_in[10];
  const float* y_ini = (const float*)d_in[11];
  float* out = (float*)d_out;

  float* ws   = (float*)d_ws;
  float* Sb   = ws;                                 // KS * GP
  float* SWe  = Sb  + (size_t)KS * GP;              // KS * PW
  float* Kaug = SWe + (size_t)KS * PW;              // NKP * WFULL  (~46 MB, L2-resident)
  float* Xg   = Kaug + (size_t)NKP * WFULL;         // GP * NBATCH

  // 1) S and weighted S|Z
  {
    int total = KS * PW;
    build_S<<<(total + 255) / 256, 256, 0, stream>>>(Yf, Uf, q, r, yref, uref, Sb, SWe);
  }
  // 2) H block + rhs columns via WMMA GEMM with fused epilogue
  {
    int waves = (GP / 16) * (PW / 64);
    gemm_H_rhs<<<(waves + 7) / 8, 256, 0, stream>>>(Sb, SWe, IPI, lam, Kaug);
  }
  // 3) remaining blocks of [K | rhs]
  {
    int total = NKP * WFULL;
    assemble_K<<<(total + 255) / 256, 256, 0, stream>>>(Up, Yp, u_ini, y_ini, Kaug);
  }
  // 4) blocked LU (no pivoting; quasi-definite KKT), rhs carried through
  for (int k = 0; k < NTK; ++k) {
    lu_diag<<<1, 256, 0, stream>>>(Kaug, k);
    int rows = NKP - (k + 1) * 16;
    if (rows > 0)
      lu_l21<<<(rows + 255) / 256, 256, 0, stream>>>(Kaug, k);
    int cols = WFULL - (k + 1) * 16;
    if (cols > 0)
      lu_u12<<<(cols + 255) / 256, 256, 0, stream>>>(Kaug, k);
    int waves = ((NTK - 1) - k) * ((WT - 1) - k);
    if (waves > 0)
      lu_trailing<<<(waves + 7) / 8, 256, 0, stream>>>(Kaug, k);
  }
  // 5) blocked back substitution
  for (int kb = NTK - 1; kb >= 0; --kb) {
    bs_diag<<<1, 128, 0, stream>>>(Kaug, kb);
    int waves = kb * (NBATCH / 16);
    if (waves > 0)
      bs_update<<<(waves + 7) / 8, 256, 0, stream>>>(Kaug, kb);
  }
  // 6) outputs: u_opt = (Uf@g).T, y_opt = (Yf@g).T
  {
    int total = GP * NBATCH;
    extract_g<<<(total + 255) / 256, 256, 0, stream>>>(Kaug, Xg);
    int wu = (320 / 16) * (NBATCH / 16);
    out_gemm<<<(wu + 7) / 8, 256, 0, stream>>>(Sb, Xg, out, /*srow0=*/480,
                                               /*MT=*/320 / 16, /*outLD=*/320);
    int wy = (480 / 16) * (NBATCH / 16);
    out_gemm<<<(wy + 7) / 8, 256, 0, stream>>>(Sb, Xg, out + NBATCH * 320,
                                               /*srow0=*/0, /*MT=*/480 / 16,
                                               /*outLD=*/480);
  }
}